// AttentiveGraph_23570780520554
// MI455X (gfx1250) — compile-verified
//
#include <hip/hip_runtime.h>
#include <cstddef>

// Problem constants (from reference)
#define B_  4
#define N_  10000
#define E_  160000
#define F_  128
#define C_  128
#define M_  (B_ * N_)          // 40000 flattened rows
#define E2_ (2 * E_)           // bidirectional edges per batch

typedef __attribute__((ext_vector_type(2))) float v2f;
typedef __attribute__((ext_vector_type(8))) float v8f;

// ---------------------------------------------------------------------------
// WMMA fp32 16x16x4 fragment layout (wave32), per CDNA5 ISA tables:
//   A 16x4 : lane -> row = lane&15, K = 4*step + (lane>>4)*2 + {0,1}  (v2f)
//   B 4x16 : lane -> col = lane&15, K = vgpr + (lane>>4)*2            (v2f)
//   C 16x16: vgpr v -> row = v + (lane>>4)*8, col = lane&15           (v8f)
// ---------------------------------------------------------------------------

// states = tanh(objects @ W + bias)   [M_ x F_] @ [F_ x C_]
__global__ void __launch_bounds__(256)
k_init_states(const float* __restrict__ obj, const float* __restrict__ W,
              const float* __restrict__ bias, float* __restrict__ out)
{
    const int lane = threadIdx.x & 31;
    const int wave = threadIdx.x >> 5;
    const int row0 = blockIdx.x * 16;
    const int col0 = wave * 16;
    const int mrow = lane & 15;
    const int half = lane >> 4;
    const int koff = half * 2;

    v8f acc = {};
    const float* arow = obj + (size_t)(row0 + mrow) * F_;
#pragma unroll 4
    for (int ks = 0; ks < F_; ks += 4) {
        v2f a = *(const v2f*)(arow + ks + koff);
        v2f b;
        b.x = W[(size_t)(ks + koff)     * C_ + col0 + mrow];
        b.y = W[(size_t)(ks + koff + 1) * C_ + col0 + mrow];
        acc = __builtin_amdgcn_wmma_f32_16x16x4_f32(false, a, false, b,
                                                    (short)0, acc, false, false);
    }
    const int col = col0 + mrow;
    const float bv = bias[col];
#pragma unroll
    for (int v = 0; v < 8; ++v) {
        const size_t idx = (size_t)(row0 + v + half * 8) * C_ + col;
        out[idx] = tanhf(acc[v] + bv);
    }
}

// aW = states @ Wa ; lW = states @ Wl   (shared A fragments, two accumulators)
__global__ void __launch_bounds__(256)
k_attn_mm(const float* __restrict__ st, const float* __restrict__ Wa,
          const float* __restrict__ Wl, float* __restrict__ aW,
          float* __restrict__ lW)
{
    const int lane = threadIdx.x & 31;
    const int wave = threadIdx.x >> 5;
    const int row0 = blockIdx.x * 16;
    const int col0 = wave * 16;
    const int mrow = lane & 15;
    const int half = lane >> 4;
    const int koff = half * 2;

    v8f acc0 = {};
    v8f acc1 = {};
    const float* arow = st + (size_t)(row0 + mrow) * C_;
#pragma unroll 4
    for (int ks = 0; ks < C_; ks += 4) {
        v2f a = *(const v2f*)(arow + ks + koff);
        v2f b0, b1;
        const size_t r0 = (size_t)(ks + koff) * C_ + col0 + mrow;
        const size_t r1 = (size_t)(ks + koff + 1) * C_ + col0 + mrow;
        b0.x = Wa[r0]; b0.y = Wa[r1];
        b1.x = Wl[r0]; b1.y = Wl[r1];
        acc0 = __builtin_amdgcn_wmma_f32_16x16x4_f32(false, a, false, b0,
                                                     (short)0, acc0, false, false);
        acc1 = __builtin_amdgcn_wmma_f32_16x16x4_f32(false, a, false, b1,
                                                     (short)0, acc1, false, false);
    }
    const int col = col0 + mrow;
#pragma unroll
    for (int v = 0; v < 8; ++v) {
        const size_t idx = (size_t)(row0 + v + half * 8) * C_ + col;
        aW[idx] = acc0[v];
        lW[idx] = acc1[v];
    }
}

// norm = 1.0, linked_gated = 0.0
__global__ void __launch_bounds__(256)
k_init_norm(float4* __restrict__ norm, float4* __restrict__ lg)
{
    const size_t i = (size_t)blockIdx.x * blockDim.x + threadIdx.x;
    if (i >= (size_t)M_ * C_ / 4) return;
    norm[i] = make_float4(1.f, 1.f, 1.f, 1.f);
    lg[i]   = make_float4(0.f, 0.f, 0.f, 0.f);
}

__device__ __forceinline__ void edge_endpoints(const int* __restrict__ conn,
                                               int b, int j, int& src, int& dst)
{
    const int* cb = conn + (size_t)b * E_ * 2;
    if (j < E_) { src = cb[2 * j];           dst = cb[2 * j + 1]; }
    else        { const int jj = j - E_; src = cb[2 * jj + 1]; dst = cb[2 * jj]; }
}

// Pass 1: norm[src] += exp(aW[src] + lW[dst] + ab)     (one wave per edge)
__global__ void __launch_bounds__(256)
k_edge_norm(const float* __restrict__ aW, const float* __restrict__ lW,
            const float* __restrict__ ab, const int* __restrict__ conn,
            float* __restrict__ norm)
{
    const int lane = threadIdx.x & 31;
    const int wave = threadIdx.x >> 5;
    const int eid  = blockIdx.x * 8 + wave;
    if (eid >= B_ * E2_) return;
    const int b = eid / E2_;
    const int j = eid % E2_;
    int src, dst;
    edge_endpoints(conn, b, j, src, dst);

    const int c = lane * 4;
    const size_t srow = ((size_t)(b * N_ + src)) * C_ + c;
    const size_t drow = ((size_t)(b * N_ + dst)) * C_ + c;
    const float4 a4 = *(const float4*)(aW + srow);
    const float4 l4 = *(const float4*)(lW + drow);
    const float4 b4 = *(const float4*)(ab + c);
    float* np = norm + srow;
    unsafeAtomicAdd(np + 0, __expf(a4.x + l4.x + b4.x));
    unsafeAtomicAdd(np + 1, __expf(a4.y + l4.y + b4.y));
    unsafeAtomicAdd(np + 2, __expf(a4.z + l4.z + b4.z));
    unsafeAtomicAdd(np + 3, __expf(a4.w + l4.w + b4.w));
}

// Pass 2: lg[src] += exp(...)/norm[src] * states[dst]  (recompute exp; L2-resident)
__global__ void __launch_bounds__(256)
k_edge_agg(const float* __restrict__ aW, const float* __restrict__ lW,
           const float* __restrict__ ab, const int* __restrict__ conn,
           const float* __restrict__ norm, const float* __restrict__ st,
           float* __restrict__ lg)
{
    const int lane = threadIdx.x & 31;
    const int wave = threadIdx.x >> 5;
    const int eid  = blockIdx.x * 8 + wave;
    if (eid >= B_ * E2_) return;
    const int b = eid / E2_;
    const int j = eid % E2_;
    int src, dst;
    edge_endpoints(conn, b, j, src, dst);

    const int c = lane * 4;
    const size_t srow = ((size_t)(b * N_ + src)) * C_ + c;
    const size_t drow = ((size_t)(b * N_ + dst)) * C_ + c;
    const float4 a4 = *(const float4*)(aW + srow);
    const float4 l4 = *(const float4*)(lW + drow);
    const float4 b4 = *(const float4*)(ab + c);
    const float4 n4 = *(const float4*)(norm + srow);
    const float4 s4 = *(const float4*)(st + drow);
    float* gp = lg + srow;
    // fast divide: v_rcp_f32 + v_mul instead of full IEEE div expansion
    unsafeAtomicAdd(gp + 0, __fdividef(__expf(a4.x + l4.x + b4.x), n4.x) * s4.x);
    unsafeAtomicAdd(gp + 1, __fdividef(__expf(a4.y + l4.y + b4.y), n4.y) * s4.y);
    unsafeAtomicAdd(gp + 2, __fdividef(__expf(a4.z + l4.z + b4.z), n4.z) * s4.z);
    unsafeAtomicAdd(gp + 3, __fdividef(__expf(a4.w + l4.w + b4.w), n4.w) * s4.w);
}

// out = tanh(st/norm + lg @ W + bias)
__global__ void __launch_bounds__(256)
k_update(const float* __restrict__ lg, const float* __restrict__ W,
         const float* __restrict__ st, const float* __restrict__ norm,
         const float* __restrict__ bias, float* __restrict__ out)
{
    const int lane = threadIdx.x & 31;
    const int wave = threadIdx.x >> 5;
    const int row0 = blockIdx.x * 16;
    const int col0 = wave * 16;
    const int mrow = lane & 15;
    const int half = lane >> 4;
    const int koff = half * 2;

    v8f acc = {};
    const float* arow = lg + (size_t)(row0 + mrow) * C_;
#pragma unroll 4
    for (int ks = 0; ks < C_; ks += 4) {
        v2f a = *(const v2f*)(arow + ks + koff);
        v2f b;
        b.x = W[(size_t)(ks + koff)     * C_ + col0 + mrow];
        b.y = W[(size_t)(ks + koff + 1) * C_ + col0 + mrow];
        acc = __builtin_amdgcn_wmma_f32_16x16x4_f32(false, a, false, b,
                                                    (short)0, acc, false, false);
    }
    const int col = col0 + mrow;
    const float bv = bias[col];
#pragma unroll
    for (int v = 0; v < 8; ++v) {
        const size_t idx = (size_t)(row0 + v + half * 8) * C_ + col;
        out[idx] = tanhf(__fdividef(st[idx], norm[idx]) + acc[v] + bv);
    }
}

extern "C" void kernel_launch(void* const* d_in, const int* in_sizes, int n_in,
                              void* d_out, int out_size, void* d_ws, size_t ws_size,
                              hipStream_t stream)
{
    const float* objects = (const float*)d_in[0];  // [B,N,F]
    const float* Wos     = (const float*)d_in[1];  // [F,C]
    const float* Wa      = (const float*)d_in[2];  // [C,C]
    const float* Wla     = (const float*)d_in[3];  // [C,C]
    const float* ab      = (const float*)d_in[4];  // [C]
    const float* Wls     = (const float*)d_in[5];  // [C,C]
    const float* sb      = (const float*)d_in[6];  // [C]
    const int*   conn    = (const int*)d_in[7];    // [B,E,2]
    float* out = (float*)d_out;                    // [B,N,C]

    float* ws = (float*)d_ws;
    const size_t SZ = (size_t)M_ * C_;             // 5.12M floats per buffer
    float* s0   = ws;
    float* s1   = ws + 1 * SZ;
    float* aW   = ws + 2 * SZ;
    float* lW   = ws + 3 * SZ;
    float* norm = ws + 4 * SZ;
    float* lg   = ws + 5 * SZ;

    const dim3 blk(256);
    const dim3 gemmGrid(M_ / 16);                  // 2500 blocks, 8 waves/blk = 128 cols
    const int edgeBlocks = (B_ * E2_ + 7) / 8;     // one wave per edge
    const int initBlocks = (int)((SZ / 4 + 255) / 256);

    k_init_states<<<gemmGrid, blk, 0, stream>>>(objects, Wos, sb, s0);

    float* cur = s0;
    float* nxt = s1;
    for (int it = 0; it < 3; ++it) {
        k_attn_mm<<<gemmGrid, blk, 0, stream>>>(cur, Wa, Wla, aW, lW);
        k_init_norm<<<initBlocks, blk, 0, stream>>>((float4*)norm, (float4*)lg);
        k_edge_norm<<<edgeBlocks, blk, 0, stream>>>(aW, lW, ab, conn, norm);
        k_edge_agg<<<edgeBlocks, blk, 0, stream>>>(aW, lW, ab, conn, norm, cur, lg);
        float* dstp = (it == 2) ? out : nxt;
        k_update<<<gemmGrid, blk, 0, stream>>>(lg, Wls, cur, norm, sb, dstp);
        float* t = cur; cur = nxt; nxt = t;
    }
}